// SchnetConv_36790689857682
// MI455X (gfx1250) — compile-verified
//
#include <hip/hip_runtime.h>
#include <hip/hip_bf16.h>
#include <math.h>

#define IN_FEATS 64
#define N_NODES  100000
#define N_EDGES  1600000

typedef __attribute__((ext_vector_type(2))) float v2f;
typedef __attribute__((ext_vector_type(8))) float v8f;

// ---------------------------------------------------------------------------
// Zero the aggregation buffer (25.6 MB of f32) each call (ws is not re-poisoned).
// ---------------------------------------------------------------------------
__global__ void zero_kernel(float* __restrict__ p, int n) {
    int i = blockIdx.x * blockDim.x + threadIdx.x;
    if (i < n) p[i] = 0.0f;
}

// ---------------------------------------------------------------------------
// Edge phase: one wave (32 lanes) per edge, 2 features per lane.
//   msg[e,k] = h[src[e],k] * exp(-gamma*(d-mu_k)^2) * fcut(d)
//   agg[dst[e],k] += msg[e,k]   (native f32 atomics; agg is L2-resident)
// ---------------------------------------------------------------------------
__global__ void edge_kernel(const float* __restrict__ h,
                            const float* __restrict__ dist,
                            const int*   __restrict__ src,
                            const int*   __restrict__ dst,
                            float*       __restrict__ agg,
                            int E) {
    int e = blockIdx.x * (blockDim.x >> 5) + (threadIdx.x >> 5);
    if (e >= E) return;
    int lane = threadIdx.x & 31;

    float d = dist[e];
    int s = src[e];
    int t = dst[e];

    // smooth cosine cutoff: onset 0.8, cutoff 1.0
    float fc;
    if (d <= 0.8f) {
        fc = 1.0f;
    } else if (d >= 1.0f) {
        fc = 0.0f;
    } else {
        float x = (d - 0.8f) * 5.0f;
        fc = 0.5f * (cosf(3.14159265358979323846f * x) + 1.0f);
    }

    int k0 = lane * 2;
    const float step = 1.0f / 63.0f;          // muk = linspace(0, 1, 64)
    float mu0 = (float)k0 * step;
    float mu1 = (float)(k0 + 1) * step;
    float d0 = d - mu0, d1 = d - mu1;
    const float gamma = 64.0f;                // IN_FEATS / (CUTOFF - 0)
    float r0 = expf(-gamma * d0 * d0) * fc;
    float r1 = expf(-gamma * d1 * d1) * fc;

    const float2 hv = *(const float2*)(h + (size_t)s * IN_FEATS + k0);
    float* ap = agg + (size_t)t * IN_FEATS + k0;
    unsafeAtomicAdd(ap,     hv.x * r0);       // -> global_atomic_add_f32
    unsafeAtomicAdd(ap + 1, hv.y * r1);
}

// ---------------------------------------------------------------------------
// Dense layer: out[M,64] = act(A[M,64] @ W[64,64] + b[64])
// One wave per 16-row M-tile; fp32 WMMA 16x16x4; all 4 N-tiles per wave.
// ISA fp32 layouts:
//   A 16x4 : lane row = lane%16;  elem0 -> K = 2*(lane/16), elem1 -> K+1
//   B 4x16 : lane col = lane%16;  elem0 -> K = 2*(lane/16), elem1 -> K+1
//   D 16x16: VGPR v -> M = v + 8*(lane/16), N = lane%16
// ---------------------------------------------------------------------------
__global__ void dense64_kernel(const float* __restrict__ A,
                               const float* __restrict__ W,
                               const float* __restrict__ bias,
                               float*       __restrict__ out,
                               int mtiles, int apply_softplus) {
    int wave = (blockIdx.x * blockDim.x + threadIdx.x) >> 5;
    if (wave >= mtiles) return;               // wave-uniform -> EXEC all-1s
    int lane  = threadIdx.x & 31;
    int row   = wave * 16 + (lane & 15);
    int khalf = lane >> 4;                    // 0 or 1

    v8f acc[4] = {};
    #pragma unroll
    for (int ks = 0; ks < 16; ++ks) {
        int kb = ks * 4 + khalf * 2;
        v2f a;
        a.x = A[(size_t)row * IN_FEATS + kb];
        a.y = A[(size_t)row * IN_FEATS + kb + 1];
        #pragma unroll
        for (int nt = 0; nt < 4; ++nt) {
            int col = nt * 16 + (lane & 15);
            v2f bf;
            bf.x = W[(size_t)(kb)     * IN_FEATS + col];
            bf.y = W[(size_t)(kb + 1) * IN_FEATS + col];
            acc[nt] = __builtin_amdgcn_wmma_f32_16x16x4_f32(
                false, a, false, bf, (short)0, acc[nt], false, false);
        }
    }

    #pragma unroll
    for (int nt = 0; nt < 4; ++nt) {
        int col = nt * 16 + (lane & 15);
        float bv = bias[col];
        #pragma unroll
        for (int v = 0; v < 8; ++v) {
            int r = wave * 16 + v + khalf * 8;
            float x = acc[nt][v] + bv;
            if (apply_softplus)               // stable softplus = logaddexp(x,0)
                x = fmaxf(x, 0.0f) + log1pf(expf(-fabsf(x)));
            out[(size_t)r * IN_FEATS + col] = x;
        }
    }
}

// ---------------------------------------------------------------------------
// Launch
// inputs: 0:h 1:dist 2:W1 3:b1 4:W2 5:b2 6:src_idx 7:dst_idx
// ws: [0, 25.6MB) agg   [25.6MB, 51.2MB) hid
// ---------------------------------------------------------------------------
extern "C" void kernel_launch(void* const* d_in, const int* in_sizes, int n_in,
                              void* d_out, int out_size, void* d_ws, size_t ws_size,
                              hipStream_t stream) {
    const float* h    = (const float*)d_in[0];
    const float* dist = (const float*)d_in[1];
    const float* W1   = (const float*)d_in[2];
    const float* b1   = (const float*)d_in[3];
    const float* W2   = (const float*)d_in[4];
    const float* b2   = (const float*)d_in[5];
    const int*   src  = (const int*)d_in[6];
    const int*   dst  = (const int*)d_in[7];

    float* agg = (float*)d_ws;
    float* hid = agg + (size_t)N_NODES * IN_FEATS;
    float* out = (float*)d_out;

    const int agg_n = N_NODES * IN_FEATS;                 // 6.4M floats
    zero_kernel<<<(agg_n + 255) / 256, 256, 0, stream>>>(agg, agg_n);

    const int waves_per_block = 8;                        // 256 threads
    int eblocks = (N_EDGES + waves_per_block - 1) / waves_per_block;
    edge_kernel<<<eblocks, 256, 0, stream>>>(h, dist, src, dst, agg, N_EDGES);

    const int mtiles = N_NODES / 16;                      // 6250 exactly
    int gblocks = (mtiles + waves_per_block - 1) / waves_per_block;
    dense64_kernel<<<gblocks, 256, 0, stream>>>(agg, W1, b1, hid, mtiles, 1);
    dense64_kernel<<<gblocks, 256, 0, stream>>>(hid, W2, b2, out, mtiles, 0);
}